// Graph_convolution_3401614098590
// MI455X (gfx1250) — compile-verified
//
#include <hip/hip_runtime.h>
#include <hip/hip_bf16.h>

typedef __attribute__((ext_vector_type(2))) float v2f;
typedef __attribute__((ext_vector_type(8))) float v8f;

#define FEAT 128
#define NK   8

// ---------------- degree / normalization ----------------
__global__ void deg_init_kernel(float* deg, int N) {
    int i = blockIdx.x * blockDim.x + threadIdx.x;
    if (i < N) deg[i] = 1.0f;   // self-loop contributes 1
}

__global__ void deg_count_kernel(const int* __restrict__ dst, float* deg, int E) {
    int i = blockIdx.x * blockDim.x + threadIdx.x;
    if (i < E) atomicAdd(&deg[dst[i]], 1.0f);
}

__global__ void dis_kernel(float* deg, int N) {
    int i = blockIdx.x * blockDim.x + threadIdx.x;
    if (i < N) deg[i] = rsqrtf(deg[i]);   // deg >= 1 always (self loops)
}

// xa[n] = dis[n]^2 * x[n]  (self-loop term of the normalized aggregation)
__global__ void xa_init_kernel(const float* __restrict__ x, const float* __restrict__ dis,
                               float* __restrict__ xa, long long total) {
    long long i = (long long)blockIdx.x * blockDim.x + threadIdx.x;
    if (i < total) {
        int n = (int)(i >> 7);
        float d = dis[n];
        xa[i] = d * d * x[i];
    }
}

// one wave per edge: xa[dst] += x[src] * dis[src]*dis[dst]
__global__ void scatter_kernel(const float* __restrict__ x,
                               const int* __restrict__ src, const int* __restrict__ dst,
                               const float* __restrict__ dis, float* __restrict__ xa, int E) {
    int wave = (blockIdx.x * blockDim.x + threadIdx.x) >> 5;
    int lane = threadIdx.x & 31;
    if (wave >= E) return;
    int s = src[wave];
    int d = dst[wave];
    float w = dis[s] * dis[d];
    const float4* xs = (const float4*)(x + (size_t)s * FEAT);
    float4 v = xs[lane];                      // 32 lanes * 4 = 128 feats
    float* o = xa + (size_t)d * FEAT + lane * 4;
    atomicAdd(o + 0, v.x * w);
    atomicAdd(o + 1, v.y * w);
    atomicAdd(o + 2, v.z * w);
    atomicAdd(o + 3, v.w * w);
}

// one wave per node: coeff[n] = softmax(x[n] @ W_dict + b_dict)
__global__ void coeff_kernel(const float* __restrict__ x, const float* __restrict__ Wd,
                             const float* __restrict__ bd, float* __restrict__ coeff, int N) {
    int node = (blockIdx.x * blockDim.x + threadIdx.x) >> 5;
    int lane = threadIdx.x & 31;
    if (node >= N) return;
    const float* xr = x + (size_t)node * FEAT;
    float xv[4];
#pragma unroll
    for (int j = 0; j < 4; ++j) xv[j] = xr[lane + 32 * j];
    float logit[NK];
#pragma unroll
    for (int a = 0; a < NK; ++a) {
        float p = 0.f;
#pragma unroll
        for (int j = 0; j < 4; ++j) p += xv[j] * Wd[(lane + 32 * j) * NK + a];
#pragma unroll
        for (int off = 16; off >= 1; off >>= 1) p += __shfl_xor(p, off, 32);
        logit[a] = p + bd[a];
    }
    float mx = logit[0];
#pragma unroll
    for (int a = 1; a < NK; ++a) mx = fmaxf(mx, logit[a]);
    float e[NK], sum = 0.f;
#pragma unroll
    for (int a = 0; a < NK; ++a) { e[a] = __expf(logit[a] - mx); sum += e[a]; }
    float inv = 1.0f / sum;
    if (lane == 0) {
#pragma unroll
        for (int a = 0; a < NK; ++a) coeff[(size_t)node * NK + a] = e[a] * inv;
    }
}

// ---------------- fused WMMA GEMM + epilogue ----------------
// total[n,c] = sum_k relu( (xa @ W[k])[n,c] + b[k,c] ) * coeff[n,k]
// One wave per 16x16 output tile; 8 fp32 accumulators (one per k).
// A rows past N are address-clamped (not zeroed): WMMA row m of D depends only
// on row m of A, and the epilogue never stores rows >= N, so garbage A rows
// are harmless. This keeps the hot loop free of per-lane exec predication.
__global__ __launch_bounds__(256)
void fused_gemm_kernel(const float* __restrict__ xa, const float* __restrict__ W,
                       const float* __restrict__ bias, const float* __restrict__ coeff,
                       float* __restrict__ out, int N, int numRowTiles) {
    int waveG = (blockIdx.x * blockDim.x + threadIdx.x) >> 5;
    int lane  = threadIdx.x & 31;
    int rowTile = waveG >> 3;          // FEAT/16 = 8 column tiles
    int colTile = waveG & 7;
    if (rowTile >= numRowTiles) return;

    int half = lane >> 4;              // 0: lanes 0-15, 1: lanes 16-31
    int l16  = lane & 15;
    int arow = rowTile * 16 + l16;     // A row held by this lane
    int bcol = colTile * 16 + l16;     // B/D column held by this lane
    int arow_c = arow < N ? arow : (N - 1);          // clamp: always a valid address
    const float* arow_ptr = xa + (size_t)arow_c * FEAT;

    v8f acc[NK] = {};                  // 8 VGPRs each, one accumulator per k

    for (int kk = 0; kk < FEAT; kk += 4) {
        int kc = kk + 2 * half;        // ISA A layout: lanes 0-15 K={0,1}, 16-31 K={2,3}
        v2f a = *(const v2f*)(arow_ptr + kc);        // aligned 8B vector load
#pragma unroll
        for (int k = 0; k < NK; ++k) {
            const float* wb = W + (size_t)k * FEAT * FEAT + (size_t)kc * FEAT + bcol;
            v2f bf;
            bf.x = wb[0];              // K = kc   row
            bf.y = wb[FEAT];           // K = kc+1 row
            acc[k] = __builtin_amdgcn_wmma_f32_16x16x4_f32(
                false, a, false, bf, (short)0, acc[k], false, false);
        }
    }

    // Epilogue: bias + relu + coeff mix, write each output element once.
#pragma unroll
    for (int r = 0; r < 8; ++r) {
        int m = r + 8 * half;          // C/D layout: VGPR r -> rows r and r+8
        int node = rowTile * 16 + m;
        if (node >= N) continue;
        float tot = 0.f;
#pragma unroll
        for (int k = 0; k < NK; ++k) {
            float v = acc[k][r] + bias[k * FEAT + bcol];
            v = v > 0.f ? v : 0.f;
            tot += v * coeff[(size_t)node * NK + k];
        }
        out[(size_t)node * FEAT + bcol] = tot;
    }
}

// ---------------- launch ----------------
extern "C" void kernel_launch(void* const* d_in, const int* in_sizes, int n_in,
                              void* d_out, int out_size, void* d_ws, size_t ws_size,
                              hipStream_t stream) {
    const float* x  = (const float*)d_in[0];
    const int*   ei = (const int*)d_in[1];
    const float* W  = (const float*)d_in[2];
    const float* b  = (const float*)d_in[3];
    const float* Wd = (const float*)d_in[4];
    const float* bd = (const float*)d_in[5];
    float* out = (float*)d_out;

    int N = in_sizes[0] / FEAT;
    int E = in_sizes[1] / 2;
    const int* src = ei;
    const int* dst = ei + E;

    // workspace layout: dis[N] | coeff[N*8] | xa[N*128]  (~27.4 MB for N=50000)
    float* dis   = (float*)d_ws;
    float* coeff = dis + N;
    float* xa    = coeff + (size_t)N * NK;

    int t = 256;
    deg_init_kernel <<<(N + t - 1) / t, t, 0, stream>>>(dis, N);
    deg_count_kernel<<<(E + t - 1) / t, t, 0, stream>>>(dst, dis, E);
    dis_kernel      <<<(N + t - 1) / t, t, 0, stream>>>(dis, N);

    long long total = (long long)N * FEAT;
    xa_init_kernel<<<(int)((total + t - 1) / t), t, 0, stream>>>(x, dis, xa, total);

    // one wave (32 threads) per edge
    long long sthreads = (long long)E * 32;
    scatter_kernel<<<(int)((sthreads + t - 1) / t), t, 0, stream>>>(x, src, dst, dis, xa, E);

    // one wave per node
    long long cthreads = (long long)N * 32;
    coeff_kernel<<<(int)((cthreads + t - 1) / t), t, 0, stream>>>(x, Wd, bd, coeff, N);

    int rowTiles = (N + 15) / 16;
    long long gthreads = (long long)rowTiles * 8 * 32;   // one wave per 16x16 tile
    fused_gemm_kernel<<<(int)((gthreads + t - 1) / t), t, 0, stream>>>(
        xa, W, b, coeff, out, N, rowTiles);
}